// VQizer_7103875908263
// MI455X (gfx1250) — compile-verified
//
#include <hip/hip_runtime.h>

typedef float v2f __attribute__((ext_vector_type(2)));
typedef float v8f __attribute__((ext_vector_type(8)));

// Problem dims (fixed by reference): B=2,S=2048,E=2048 ; H=16,O=1024,A=128
#define E_DIM 2048
#define H_DIM 16
#define O_DIM 1024
#define A_DIM 128
#define TOK_PER_BLK 16

// Monotone map f32 -> u32 so unsigned compare == float compare (no NaNs expected).
__device__ __forceinline__ unsigned ord32(float f) {
    unsigned u = __float_as_uint(f);
    return (u & 0x80000000u) ? ~u : (u | 0x80000000u);
}

__global__ __launch_bounds__(256)
void vq_argmax_gather_kernel(const float* __restrict__ x,
                             const float* __restrict__ w,
                             const float* __restrict__ cb,
                             float* __restrict__ out)
{
    __shared__ unsigned long long smax[TOK_PER_BLK];

    const int head     = blockIdx.x & (H_DIM - 1);
    const int tok_base = (blockIdx.x >> 4) * TOK_PER_BLK;   // 256 token blocks
    const int tid      = threadIdx.x;
    const int lane     = tid & 31;
    const int wave     = tid >> 5;          // 8 waves
    const int lane16   = lane & 15;
    const int khalf    = (lane >> 4) << 1;  // 0 for lanes 0-15, 2 for lanes 16-31

    if (tid < TOK_PER_BLK) smax[tid] = 0ull;
    __syncthreads();

    // ---- Preload A block into registers: x[tok_base+lane16][head*128 + 4i + khalf .. +1]
    // Matches 32-bit A 16x4 WMMA layout: VGPR0/1 = K=k0,k0+1 (lanes 0-15), K=k0+2,k0+3 (lanes 16-31).
    const float* xrow = x + (size_t)(tok_base + lane16) * E_DIM + head * A_DIM + khalf;
    v2f a[32];
#pragma unroll
    for (int i = 0; i < 32; ++i)
        a[i] = *(const v2f*)(xrow + 4 * i);

    float vmax[8];
    int   imax[8];
#pragma unroll
    for (int r = 0; r < 8; ++r) { vmax[r] = -__builtin_huge_valf(); imax[r] = 0; }

    const int    o_wave = wave * 128;                 // each wave owns 128 options
    const float* wbase  = w + (size_t)head * O_DIM * A_DIM;

#pragma unroll 1
    for (int tile = 0; tile < 8; ++tile) {
        const int    o0   = o_wave + tile * 16;
        // B layout (4x16, K x N): lane n=lane16 holds B[k0+khalf][n], B[k0+1+khalf][n]
        // B[k][n] = W[o0+n][k]  -> contiguous float2 along k in W's row.
        const float* wrow = wbase + (size_t)(o0 + lane16) * A_DIM + khalf;
        v8f acc = {};
#pragma unroll
        for (int i = 0; i < 32; ++i) {
            v2f b = *(const v2f*)(wrow + 4 * i);
            acc = __builtin_amdgcn_wmma_f32_16x16x4_f32(
                false, a[i], false, b, (short)0, acc, false, false);
        }
        // C/D layout: lane j<16: VGPR r -> (M=r, N=j); lane j>=16: (M=8+r, N=j-16).
        const int oidx = o0 + lane16;
#pragma unroll
        for (int r = 0; r < 8; ++r) {
            if (acc[r] > vmax[r]) { vmax[r] = acc[r]; imax[r] = oidx; }  // strict > keeps lowest idx
        }
    }

    // ---- Cross-lane / cross-wave argmax: packed (ordered float, ~idx) u64 max.
    // Max picks largest float; among equals, largest ~idx == smallest idx (JAX tie-break).
#pragma unroll
    for (int r = 0; r < 8; ++r) {
        const int m = r + ((lane >> 4) << 3);   // token row this lane's VGPR r belongs to
        unsigned long long key =
            ((unsigned long long)ord32(vmax[r]) << 32) | (unsigned)(~(unsigned)imax[r]);
        atomicMax(&smax[m], key);               // ds_max_u64
    }
    __syncthreads();

    // ---- Gather codebook rows: 16 tokens x 128 floats, 256 threads x 8 floats.
    const int t   = tid >> 4;            // token 0..15
    const int ac8 = (tid & 15) * 8;      // a-offset
    const unsigned long long k = smax[t];
    const int idx = (int)(~(unsigned)(k & 0xFFFFFFFFull));

    const float4* src = (const float4*)(cb + ((size_t)head * O_DIM + idx) * A_DIM + ac8);
    float4*       dst = (float4*)(out + (size_t)(tok_base + t) * E_DIM + head * A_DIM + ac8);
    dst[0] = src[0];
    dst[1] = src[1];
}

extern "C" void kernel_launch(void* const* d_in, const int* in_sizes, int n_in,
                              void* d_out, int out_size, void* d_ws, size_t ws_size,
                              hipStream_t stream) {
    (void)in_sizes; (void)n_in; (void)out_size; (void)d_ws; (void)ws_size;
    const float* x  = (const float*)d_in[0];   // [2,2048,2048]
    const float* w  = (const float*)d_in[1];   // [16,1024,128]
    const float* cb = (const float*)d_in[2];   // [16,1024,128]
    // d_in[3] = temperature: cancels out of the forward value, unused.
    float* out = (float*)d_out;                // [2,2048,2048]

    const int token_blocks = (2 * 2048) / TOK_PER_BLK;   // 256
    dim3 grid(token_blocks * H_DIM);                     // 4096 workgroups
    vq_argmax_gather_kernel<<<grid, 256, 0, stream>>>(x, w, cb, out);
}